// MixtralRouter_47029891891650
// MI455X (gfx1250) — compile-verified
//
#include <hip/hip_runtime.h>
#include <hip/hip_bf16.h>
#include <math.h>
#include <stdint.h>

// Mixtral router for MI455X (gfx1250, wave32).
// logits = x[T,H] @ W[E,H]^T via V_WMMA_F32_16X16X4_F32.
// x is streamed through LDS with GLOBAL_LOAD_ASYNC_TO_LDS_B128 (coalesced,
// double-buffered, ASYNCcnt-tracked); W is pre-swizzled into a packed WMMA
// B-fragment layout in d_ws so one coalesced global_load_b128 feeds two WMMAs.
// Memory bound: 256 MiB of x @ 23.3 TB/s ~= 11 us.

typedef float v2f __attribute__((ext_vector_type(2)));
typedef float v4f __attribute__((ext_vector_type(4)));
typedef float v8f __attribute__((ext_vector_type(8)));

#define H_DIM 4096
#define E_DIM 8
#define TOK_PER_BLK 16
#define WAVES_PER_BLK 8
#define KSLICE (H_DIM / WAVES_PER_BLK)   // 512 k per wave
#define CHUNK_K 128                      // k per staged chunk
#define NSTEP (CHUNK_K / 4)              // 32 wmma per chunk
#define XPITCH (CHUNK_K + 4)             // 132 floats -> conflict-free ds_load_b64

// ---------------- prep kernel: W -> packed WMMA B-fragment layout -----------
// B (4x16 per k-step) lane layout: lane l = (sub=l%16, half=l/16):
//   step k0: v0 = W[sub][4*k0 + 2*half], v1 = W[sub][4*k0 + 2*half + 1]
//   (zero for sub>=8 -> expert padding baked in).
// Packed pairs: p = k0/2, bf[p*128 + lane*4 + {0,1}] = step 2p,
//                         bf[p*128 + lane*4 + {2,3}] = step 2p+1.
__global__ __launch_bounds__(256)
void prep_bfrag_kernel(const float* __restrict__ W, float* __restrict__ bf) {
    const int idx  = blockIdx.x * blockDim.x + threadIdx.x;  // 0 .. 16383
    const int p    = idx >> 5;        // k-step pair index, 0..511
    const int lane = idx & 31;
    const int sub  = lane & 15;
    const int half = lane >> 4;
    v4f v = {0.0f, 0.0f, 0.0f, 0.0f};
    if (sub < E_DIM) {
        const float* wr = W + (size_t)sub * H_DIM + 8 * p + 2 * half;
        v[0] = wr[0];
        v[1] = wr[1];
        v[2] = wr[4];
        v[3] = wr[5];
    }
    *(v4f*)(bf + (size_t)idx * 4) = v;   // coalesced b128 store
}

// ---------------- shared epilogue: reduce tiles, softmax, top-2 -------------
__device__ __forceinline__
void router_epilogue(v8f acc, float (*part)[256], float (*logits_s)[E_DIM],
                     float* __restrict__ out, int T, int tb,
                     int tid, int lane, int wave) {
    #pragma unroll
    for (int v = 0; v < 8; ++v) part[wave][v * 32 + lane] = acc[v];
    __syncthreads();

    {
        float s = 0.0f;
        #pragma unroll
        for (int w = 0; w < WAVES_PER_BLK; ++w) s += part[w][tid];
        // D layout: VGPR v, lanes 0-15: M=v, N=lane; lanes 16-31: M=v+8, N=lane-16
        const int v = tid >> 5;
        const int l = tid & 31;
        const int M = v + 8 * (l >> 4);
        const int N = l & 15;
        if (N < E_DIM) logits_s[M][N] = s;
    }
    __syncthreads();

    if (tid < TOK_PER_BLK) {
        const int token = tb + tid;
        float lg[E_DIM];
        float mx = -INFINITY;
        #pragma unroll
        for (int e = 0; e < E_DIM; ++e) { lg[e] = logits_s[tid][e]; mx = fmaxf(mx, lg[e]); }
        float sum = 0.0f, sc[E_DIM];
        #pragma unroll
        for (int e = 0; e < E_DIM; ++e) { sc[e] = expf(lg[e] - mx); sum += sc[e]; }
        const float inv = 1.0f / sum;
        float* scores = out + (size_t)token * E_DIM;
        #pragma unroll
        for (int e = 0; e < E_DIM; ++e) { sc[e] *= inv; scores[e] = sc[e]; }

        // top-2, ties toward lower index (matches jax.lax.top_k)
        int i0 = 0; float v0 = sc[0];
        #pragma unroll
        for (int e = 1; e < E_DIM; ++e) { if (sc[e] > v0) { v0 = sc[e]; i0 = e; } }
        int i1 = (i0 == 0) ? 1 : 0; float v1 = sc[i1];
        #pragma unroll
        for (int e = 0; e < E_DIM; ++e) { if (e != i0 && sc[e] > v1) { v1 = sc[e]; i1 = e; } }
        const float rs = 1.0f / (v0 + v1);

        const size_t Toff = (size_t)T;
        float* wout = out + Toff * E_DIM + (size_t)token * 2;
        wout[0] = v0 * rs;
        wout[1] = v1 * rs;
        float* iout = out + Toff * E_DIM + Toff * 2 + (size_t)token * 2;
        iout[0] = (float)i0;
        iout[1] = (float)i1;
    }
}

// ---------------- main kernel: async-LDS staged WMMA ------------------------
#define WAIT_ASYNC16() asm volatile("s_wait_asynccnt 0x10" ::: "memory")
#define WAIT_ASYNC0()  asm volatile("s_wait_asynccnt 0x0"  ::: "memory")
#define WAIT_DS0()     asm volatile("s_wait_dscnt 0x0"     ::: "memory")

__global__ __launch_bounds__(256)
void router_wmma_async_kernel(const float* __restrict__ x,
                              const float* __restrict__ bf,
                              float* __restrict__ out,
                              int T) {
    // per-wave double-buffered x staging: 8*2*16*132*4 B = 132 KiB
    __shared__ __align__(16) float xs[WAVES_PER_BLK][2][TOK_PER_BLK][XPITCH];
    __shared__ float part[WAVES_PER_BLK][256];
    __shared__ float logits_s[TOK_PER_BLK][E_DIM];

    const int tid  = threadIdx.x;
    const int lane = tid & 31;
    const int wave = tid >> 5;
    const int half = lane >> 4;
    const int sub  = lane & 15;

    const int tb    = blockIdx.x * TOK_PER_BLK;
    const int kbase = wave * KSLICE;           // this wave's k slice
    const int pbase0 = (kbase >> 3);           // packed B pair-index base

    const float* xblk = x + (size_t)tb * H_DIM;

    // One async b128 per row: 32 lanes x 16B = 512B fully coalesced.
    #define ISSUE_CHUNK(BUF, CIDX)                                               \
        { const int kc_ = kbase + (CIDX) * CHUNK_K;                              \
          _Pragma("unroll")                                                      \
          for (int r = 0; r < TOK_PER_BLK; ++r) {                                \
              const void* g_ = (const void*)(xblk + (size_t)r * H_DIM + kc_ + lane * 4); \
              unsigned l_ = (unsigned)(uintptr_t)&xs[wave][BUF][r][lane * 4];    \
              asm volatile("global_load_async_to_lds_b128 %0, %1, off"           \
                           :: "v"(l_), "v"(g_) : "memory");                      \
          } }

    // A from LDS (conflict-free: banks 4*sub+2*half+{0,1} all distinct),
    // B from the packed fragment stream: one b128 feeds two WMMAs.
    #define COMPUTE_CHUNK(BUF, CIDX)                                             \
        { const int pb_ = pbase0 + (CIDX) * (NSTEP / 2);                         \
          _Pragma("unroll")                                                      \
          for (int k = 0; k < CHUNK_K; k += 8) {                                 \
              v4f b4_ = *(const v4f*)(bf + ((size_t)(pb_ + (k >> 3)) * 128 + lane * 4)); \
              v2f a0_ = *(const v2f*)&xs[wave][BUF][sub][k + 2 * half];          \
              v2f a1_ = *(const v2f*)&xs[wave][BUF][sub][k + 4 + 2 * half];      \
              v2f b0_ = __builtin_shufflevector(b4_, b4_, 0, 1);                 \
              v2f b1_ = __builtin_shufflevector(b4_, b4_, 2, 3);                 \
              acc = __builtin_amdgcn_wmma_f32_16x16x4_f32(                       \
                  false, a0_, false, b0_, (short)0, acc, false, false);          \
              acc = __builtin_amdgcn_wmma_f32_16x16x4_f32(                       \
                  false, a1_, false, b1_, (short)0, acc, false, false);          \
          } }

    v8f acc = {};

    ISSUE_CHUNK(0, 0);
    ISSUE_CHUNK(1, 1);

    WAIT_ASYNC16(); COMPUTE_CHUNK(0, 0); WAIT_DS0(); ISSUE_CHUNK(0, 2);
    WAIT_ASYNC16(); COMPUTE_CHUNK(1, 1); WAIT_DS0(); ISSUE_CHUNK(1, 3);
    WAIT_ASYNC16(); COMPUTE_CHUNK(0, 2);
    WAIT_ASYNC0();  COMPUTE_CHUNK(1, 3);

    #undef ISSUE_CHUNK
    #undef COMPUTE_CHUNK

    router_epilogue(acc, part, logits_s, out, T, tb, tid, lane, wave);
}

// ---------------- fallback (no workspace): direct global loads --------------
__global__ __launch_bounds__(256)
void router_wmma_fallback_kernel(const float* __restrict__ x,
                                 const float* __restrict__ W,
                                 float* __restrict__ out,
                                 int T) {
    __shared__ float part[WAVES_PER_BLK][256];
    __shared__ float logits_s[TOK_PER_BLK][E_DIM];

    const int tid  = threadIdx.x;
    const int lane = tid & 31;
    const int wave = tid >> 5;
    const int half = lane >> 4;
    const int sub  = lane & 15;
    const int tb   = blockIdx.x * TOK_PER_BLK;

    const float* xrow = x + (size_t)(tb + sub) * H_DIM;
    const float* wrow = W + (size_t)((sub < E_DIM) ? sub : 0) * H_DIM;
    const float  bmask = (sub < E_DIM) ? 1.0f : 0.0f;

    v8f acc = {};
    const int kbase = wave * KSLICE + 2 * half;

    #pragma unroll 4
    for (int k = 0; k < KSLICE; k += 4) {
        const int kk = kbase + k;
        v2f a = *(const v2f*)(xrow + kk);
        v2f b = *(const v2f*)(wrow + kk);
        b *= bmask;
        acc = __builtin_amdgcn_wmma_f32_16x16x4_f32(
            false, a, false, b, (short)0, acc, false, false);
    }

    router_epilogue(acc, part, logits_s, out, T, tb, tid, lane, wave);
}

// ---------------- host launch ----------------------------------------------
extern "C" void kernel_launch(void* const* d_in, const int* in_sizes, int n_in,
                              void* d_out, int out_size, void* d_ws, size_t ws_size,
                              hipStream_t stream) {
    const float* x = (const float*)d_in[0];   // [4,4096,4096] fp32
    const float* W = (const float*)d_in[1];   // [8,4096] fp32
    float* out = (float*)d_out;               // scores | weights | indices (flat)

    const int T = in_sizes[0] / H_DIM;        // 16384 tokens
    const int grid = T / TOK_PER_BLK;         // 1024 blocks

    const size_t bf_bytes = (size_t)(H_DIM / 8) * 128 * sizeof(float); // 256 KiB

    if (d_ws != nullptr && ws_size >= bf_bytes) {
        float* bf = (float*)d_ws;
        prep_bfrag_kernel<<<(H_DIM / 8) * 32 / 256, 256, 0, stream>>>(W, bf);
        router_wmma_async_kernel<<<grid, 256, 0, stream>>>(x, bf, out, T);
    } else {
        router_wmma_fallback_kernel<<<grid, 256, 0, stream>>>(x, W, out, T);
    }
}